// Attntopo_81827716923658
// MI455X (gfx1250) — compile-verified
//
#include <hip/hip_runtime.h>

typedef __attribute__((ext_vector_type(2))) float v2f;
typedef __attribute__((ext_vector_type(8))) float v8f;

#define FIN  256
#define FOUT 128
#define BPITCH 68   // LDS pitch (dwords) for transposed W tile: 68 % 64 == 4 -> conflict-free b64 reads

// -------------------------------------------------------------------------
// Kernel 1: h = x @ W via V_WMMA_F32_16X16X4_F32.
// Block = 256 threads = 8 waves; each wave owns a 16x128 tile of h.
// W is staged in LDS in 64x128 K-tiles, stored TRANSPOSED+PADDED so each
// B fragment (K,K+1 for one column) is a single aligned ds_load_b64.
// -------------------------------------------------------------------------
__global__ void __launch_bounds__(256) gat_gemm(const float* __restrict__ x,
                                                const float* __restrict__ W,
                                                float* __restrict__ h,
                                                int nrows) {
    __shared__ __align__(16) float WtT[FOUT * BPITCH];  // 34 KB: WtT[n*68 + k]

    const int lane  = threadIdx.x & 31;
    const int wave  = threadIdx.x >> 5;                 // 0..7
    const int row0  = (blockIdx.x * 8 + wave) * 16;     // 16-row M-tile origin
    const bool active = row0 < nrows;                   // wave-uniform -> EXEC stays full for WMMA

    const int n     = lane & 15;                        // A row / B,C column within tile
    const int khalf = (lane >> 4) << 1;                 // 0 or 2: per-half-wave K sub-offset

    v8f acc[8];
    #pragma unroll
    for (int t = 0; t < 8; ++t) acc[t] = (v8f){0.f,0.f,0.f,0.f,0.f,0.f,0.f,0.f};

    const float* xrow = x + (size_t)(row0 + n) * FIN + khalf;

    for (int kb = 0; kb < 4; ++kb) {                    // four 64-wide K tiles
        // cooperative transposed load: WtT[n*BPITCH + k] = W[kb*64 + k][n]
        // global reads coalesced over n; one-time 2-way LDS store conflicts are fine
        for (int i = threadIdx.x; i < 64 * FOUT; i += 256) {
            const int k = i >> 7;           // 0..63
            const int c = i & 127;          // 0..127
            WtT[c * BPITCH + k] = W[(size_t)(kb * 64 + k) * FOUT + c];
        }
        __syncthreads();

        if (active) {
            const float* xk = xrow + kb * 64;
            if (kb < 3)                      // global_prefetch_b8: pull next K slice of x
                __builtin_prefetch(xk + 64, 0, 1);
            for (int kk = 0; kk < 16; ++kk) {            // 16 k-steps of 4
                const int k = kk * 4 + khalf;            // even -> 8B aligned LDS reads
                const float2 av = *(const float2*)(xk + kk * 4);
                v2f a; a[0] = av.x; a[1] = av.y;
                #pragma unroll
                for (int nt = 0; nt < 8; ++nt) {         // 8 column tiles of 16
                    const v2f b = *(const v2f*)(WtT + (nt * 16 + n) * BPITCH + k);
                    acc[nt] = __builtin_amdgcn_wmma_f32_16x16x4_f32(
                        false, a, false, b, (short)0, acc[nt], false, false);
                }
            }
        }
        __syncthreads();
    }

    if (active) {
        // C/D layout: VGPR i, lanes 0-15 -> M=i, lanes 16-31 -> M=i+8, N=lane%16
        const int rbase = row0 + ((lane >> 4) << 3);
        #pragma unroll
        for (int nt = 0; nt < 8; ++nt)
            #pragma unroll
            for (int i = 0; i < 8; ++i)
                h[(size_t)(rbase + i) * FOUT + nt * 16 + n] = acc[nt][i];
    }
}

// -------------------------------------------------------------------------
// Kernel 2: h'[row[e]] += h[col[e]]  (att == 1 after size-1 softmax).
// One wave per edge: 32 lanes x float4 = 128 floats. Native f32 fadd atomics
// via inline asm (one 64-bit address VGPR pair, immediate offsets 0/4/8/12).
// h (51 MB) and h' (51 MB) are L2-resident on MI455X (192 MB L2), so this
// phase runs at L2/atomic throughput, not HBM.
// -------------------------------------------------------------------------
__global__ void __launch_bounds__(256) gat_scatter(const long long* __restrict__ ei,
                                                   const float* __restrict__ h,
                                                   float* __restrict__ hp,
                                                   int E) {
    const int lane = threadIdx.x & 31;
    int w        = (int)((blockIdx.x * blockDim.x + threadIdx.x) >> 5);
    const int nw = (int)((gridDim.x * blockDim.x) >> 5);
    for (int e = w; e < E; e += nw) {
        const int r = (int)ei[e];                    // row = edge_index[0][e]
        const int c = (int)ei[(size_t)E + e];        // col = edge_index[1][e]
        const float4 v = *(const float4*)(h + (size_t)c * FOUT + lane * 4);
        float* p = hp + (size_t)r * FOUT + lane * 4;
        asm volatile(
            "global_atomic_add_f32 %0, %1, off\n\t"
            "global_atomic_add_f32 %0, %2, off offset:4\n\t"
            "global_atomic_add_f32 %0, %3, off offset:8\n\t"
            "global_atomic_add_f32 %0, %4, off offset:12"
            :: "v"(p), "v"(v.x), "v"(v.y), "v"(v.z), "v"(v.w)
            : "memory");
    }
}

// -------------------------------------------------------------------------
// Kernel 3: out = elu(h')  (alpha = 1)
// -------------------------------------------------------------------------
__global__ void __launch_bounds__(256) gat_elu(const float* __restrict__ hp,
                                               float* __restrict__ out,
                                               long long n) {
    long long i4 = ((long long)blockIdx.x * blockDim.x + threadIdx.x) * 4;
    if (i4 + 3 < n) {
        float4 v = *(const float4*)(hp + i4);
        v.x = v.x > 0.f ? v.x : __expf(v.x) - 1.f;
        v.y = v.y > 0.f ? v.y : __expf(v.y) - 1.f;
        v.z = v.z > 0.f ? v.z : __expf(v.z) - 1.f;
        v.w = v.w > 0.f ? v.w : __expf(v.w) - 1.f;
        *(float4*)(out + i4) = v;
    } else {
        for (long long i = i4; i < n; ++i) {
            float s = hp[i];
            out[i] = s > 0.f ? s : __expf(s) - 1.f;
        }
    }
}

// -------------------------------------------------------------------------
extern "C" void kernel_launch(void* const* d_in, const int* in_sizes, int n_in,
                              void* d_out, int out_size, void* d_ws, size_t ws_size,
                              hipStream_t stream) {
    const float*     x  = (const float*)d_in[0];      // [N, FIN]
    const long long* ei = (const long long*)d_in[1];  // [2, E] int64
    const float*     W  = (const float*)d_in[2];      // [FIN, FOUT]
    // d_in[3] ("a") is mathematically unused: softmax over a size-1 axis == 1.

    const int N = in_sizes[0] / FIN;
    const int E = in_sizes[1] / 2;

    float* h  = (float*)d_out;   // h = x @ W lives in d_out until the ELU pass
    float* hp = (float*)d_ws;    // h' accumulator (N*FOUT*4 = 51.2 MB of scratch)

    hipMemsetAsync(hp, 0, (size_t)N * FOUT * sizeof(float), stream);

    const int mtiles = (N + 15) / 16;
    gat_gemm<<<(mtiles + 7) / 8, 256, 0, stream>>>(x, W, h, N);

    gat_scatter<<<2048, 256, 0, stream>>>(ei, h, hp, E);

    const long long n = (long long)N * FOUT;
    gat_elu<<<(int)((n / 4 + 255) / 256), 256, 0, stream>>>(hp, (float*)d_out, n);
}